// GraphSAGE_1090921693773
// MI455X (gfx1250) — compile-verified
//
#include <hip/hip_runtime.h>

#define NN 50000
#define NE 800000
#define DHID 128
#define DOUT 64

typedef float v2f __attribute__((ext_vector_type(2)));
typedef float v8f __attribute__((ext_vector_type(8)));

// Guaranteed CDNA5 hardware f32 atomic (no CAS fallback). Fire-and-forget:
// tracked by STOREcnt, drained by the implicit wait-idle at S_ENDPGM.
__device__ __forceinline__ void atomic_fadd(float* p, float v) {
    asm volatile("global_atomic_add_f32 %0, %1, off" : : "v"(p), "v"(v) : "memory");
}

// ---------------- utility: zero a region with float4 stores ----------------
__global__ void zero_f4(float4* __restrict__ p, int n4) {
    int i = blockIdx.x * blockDim.x + threadIdx.x;
    if (i < n4) p[i] = make_float4(0.f, 0.f, 0.f, 0.f);
}

// ---------------- degree: one atomic per edge ----------------
__global__ void degree_kernel(const int* __restrict__ dst, float* __restrict__ cnt, int E) {
    int e = blockIdx.x * blockDim.x + threadIdx.x;
    if (e < E) atomic_fadd(&cnt[dst[e]], 1.0f);
}

// ---------------- scatter-add: one wave32 per edge, 4 floats/lane ----------------
__global__ void scatter_kernel(const int* __restrict__ src, const int* __restrict__ dst,
                               const float* __restrict__ feat, float* __restrict__ agg,
                               int E) {
    int gid  = blockIdx.x * blockDim.x + threadIdx.x;
    int e    = gid >> 5;          // one wave32 per edge
    int lane = gid & 31;
    if (e >= E) return;
    int s = src[e];
    int d = dst[e];
    const float4 v = *(const float4*)(feat + (size_t)s * DHID + lane * 4);  // coalesced 512B/row
    float* o = agg + (size_t)d * DHID + lane * 4;
    atomic_fadd(o + 0, v.x);
    atomic_fadd(o + 1, v.y);
    atomic_fadd(o + 2, v.z);
    atomic_fadd(o + 3, v.w);
}

// ---------------- fused SAGE linear: out = relu?(mean@Wl + b + h@Wr) ----------------
// One wave32 computes an 80x32 output block (MB=5 x NB=2 tiles of 16x16) with
// V_WMMA_F32_16X16X4_F32 (exact f32, K step 4). B fragments reused 5x, A 2x.
// All A fragments of a k-step are loaded into distinct registers up front so the
// loads issue as one clause instead of serializing on a recycled VGPR pair.
// A (16x4 f32): lanes 0-15 -> M=lane, K=k+0,k+1 ; lanes 16-31 -> same rows, K=k+2,k+3
// B (4x16 f32): symmetric (N in lane, K pair by lane>>4)
// C/D (16x16 f32): VGPR v, lane L -> M = v + 8*(L>>4), N = L&15
template <int LDB>   // LDB = output columns = row stride of Wl/Wr and out
__global__ void sage_gemm(const float* __restrict__ agg, const float* __restrict__ cnt,
                          const float* __restrict__ hroot,
                          const float* __restrict__ Wl, const float* __restrict__ bias,
                          const float* __restrict__ Wr,
                          float* __restrict__ out, int nrows, int relu) {
    constexpr int MB = 5;                 // 50000 / (5*16) = 625 exactly
    constexpr int NB = 2;                 // 128/32=4, 64/32=2 exactly
    const int lane = threadIdx.x & 31;
    const int wave = (blockIdx.x * blockDim.x + threadIdx.x) >> 5;
    const int ntn  = LDB / (16 * NB);
    const int stm  = wave / ntn;
    const int stn  = wave % ntn;
    if (stm * (16 * MB) >= nrows) return;

    const int half = lane >> 4;           // 0 or 1
    const int mn   = lane & 15;
    const int koff = half * 2;

    const int r0 = stm * (16 * MB) + mn;  // base A row this lane loads
    const int c0 = stn * (16 * NB) + mn;  // base B/output column this lane handles

    float rinv[MB];
#pragma unroll
    for (int mi = 0; mi < MB; ++mi)
        rinv[mi] = 1.0f / fmaxf(cnt[r0 + mi * 16], 1.0f);

    const float* ap = agg   + (size_t)r0 * DHID;   // row steps are 8KB immediate offsets
    const float* xp = hroot + (size_t)r0 * DHID;

    v8f acc[MB][NB];
#pragma unroll
    for (int mi = 0; mi < MB; ++mi)
#pragma unroll
        for (int nj = 0; nj < NB; ++nj)
            acc[mi][nj] = (v8f){0.f, 0.f, 0.f, 0.f, 0.f, 0.f, 0.f, 0.f};

#pragma unroll 2
    for (int k = 0; k < DHID; k += 4) {
        // ---- issue ALL fragment loads for this k-step first (clauseable) ----
        v2f bl[NB], br[NB];
#pragma unroll
        for (int nj = 0; nj < NB; ++nj) {
            bl[nj][0] = Wl[(k + koff)     * LDB + c0 + nj * 16];
            bl[nj][1] = Wl[(k + koff + 1) * LDB + c0 + nj * 16];
            br[nj][0] = Wr[(k + koff)     * LDB + c0 + nj * 16];
            br[nj][1] = Wr[(k + koff + 1) * LDB + c0 + nj * 16];
        }
        v2f am[MB], ar[MB];
#pragma unroll
        for (int mi = 0; mi < MB; ++mi) {
            float2 av = *(const float2*)(ap + mi * 16 * DHID + k + koff);
            am[mi][0] = av.x * rinv[mi];
            am[mi][1] = av.y * rinv[mi];
            float2 xv = *(const float2*)(xp + mi * 16 * DHID + k + koff);
            ar[mi][0] = xv.x;
            ar[mi][1] = xv.y;
        }
        // ---- matrix math: 20 WMMAs on pre-loaded fragments ----
#pragma unroll
        for (int mi = 0; mi < MB; ++mi) {
#pragma unroll
            for (int nj = 0; nj < NB; ++nj) {
                acc[mi][nj] = __builtin_amdgcn_wmma_f32_16x16x4_f32(
                    false, am[mi], false, bl[nj], (short)0, acc[mi][nj], false, false);
                acc[mi][nj] = __builtin_amdgcn_wmma_f32_16x16x4_f32(
                    false, ar[mi], false, br[nj], (short)0, acc[mi][nj], false, false);
            }
        }
    }

#pragma unroll
    for (int nj = 0; nj < NB; ++nj) {
        const float bv = bias[c0 + nj * 16];
#pragma unroll
        for (int mi = 0; mi < MB; ++mi) {
#pragma unroll
            for (int v = 0; v < 8; ++v) {
                int m = stm * (16 * MB) + mi * 16 + v + half * 8;
                float val = acc[mi][nj][v] + bv;
                if (relu) val = fmaxf(val, 0.f);
                out[(size_t)m * LDB + c0 + nj * 16] = val;
            }
        }
    }
}

// ---------------- launcher ----------------
extern "C" void kernel_launch(void* const* d_in, const int* in_sizes, int n_in,
                              void* d_out, int out_size, void* d_ws, size_t ws_size,
                              hipStream_t stream) {
    const float* x   = (const float*)d_in[0];
    const int*   ei  = (const int*)d_in[1];     // [2*E]: row0=src, row1=dst
    const float* W1l = (const float*)d_in[2];
    const float* b1  = (const float*)d_in[3];
    const float* W1r = (const float*)d_in[4];
    const float* W2l = (const float*)d_in[5];
    const float* b2  = (const float*)d_in[6];
    const float* W2r = (const float*)d_in[7];
    float*       out = (float*)d_out;

    const int* src = ei;
    const int* dst = ei + NE;

    // workspace layout: agg[NN*128] | cnt[NN] | h[NN*128]  (~51.4 MB)
    float* agg = (float*)d_ws;
    float* cnt = agg + (size_t)NN * DHID;
    float* h   = cnt + NN;

    const int BLK = 256;

    // ---- zero agg + cnt (contiguous, divisible by 4 floats) ----
    {
        int n4 = (NN * DHID + NN) / 4;
        zero_f4<<<(n4 + BLK - 1) / BLK, BLK, 0, stream>>>((float4*)agg, n4);
    }

    // ---- degrees (identical for both layers) ----
    degree_kernel<<<(NE + BLK - 1) / BLK, BLK, 0, stream>>>(dst, cnt, NE);

    // ---- layer 1: scatter x, fused WMMA GEMM -> h (ReLU) ----
    {
        long long thr = (long long)NE * 32;
        scatter_kernel<<<(int)((thr + BLK - 1) / BLK), BLK, 0, stream>>>(src, dst, x, agg, NE);

        int waves = (NN / 80) * (DHID / 32);          // 625 * 4 = 2500
        int thr2  = waves * 32;
        sage_gemm<DHID><<<(thr2 + BLK - 1) / BLK, BLK, 0, stream>>>(
            agg, cnt, x, W1l, b1, W1r, h, NN, /*relu=*/1);
    }

    // ---- layer 2: re-zero agg, scatter h, fused WMMA GEMM -> out ----
    {
        int n4 = (NN * DHID) / 4;
        zero_f4<<<(n4 + BLK - 1) / BLK, BLK, 0, stream>>>((float4*)agg, n4);

        long long thr = (long long)NE * 32;
        scatter_kernel<<<(int)((thr + BLK - 1) / BLK), BLK, 0, stream>>>(src, dst, h, agg, NE);

        int waves = (NN / 80) * (DOUT / 32);          // 625 * 2 = 1250
        int thr2  = waves * 32;
        sage_gemm<DOUT><<<(thr2 + BLK - 1) / BLK, BLK, 0, stream>>>(
            agg, cnt, h, W2l, b2, W2r, out, NN, /*relu=*/0);
    }
}